// SelectiveSSMBlock_78675210928143
// MI455X (gfx1250) — compile-verified
//
#include <hip/hip_runtime.h>
#include <hip/hip_bf16.h>

// ---------------------------------------------------------------------------
// Selective SSM (Mamba) block for MI455X / gfx1250.
// wave32 + v_wmma_f32_16x16x4_f32 GEMMs + async global->LDS (ASYNCcnt) staging.
// ---------------------------------------------------------------------------

#define D_MODEL 768
#define D_INNER 1536
#define D_STATE 16
#define D_CONV  4
#define B_SZ    4
#define T_LEN   2048
#define BT      (B_SZ * T_LEN)        // 8192
#define SP_STRIDE 40                  // padded row stride for (dt,B,C) = 33

typedef __attribute__((ext_vector_type(2))) float v2f;
typedef __attribute__((ext_vector_type(8))) float v8f;

// Low 32 bits of a flat pointer into LDS == LDS byte offset (flat->LDS mapping
// truncates addr[63:32]); used as the VDST operand of async-to-LDS loads.
__device__ __forceinline__ unsigned lds_u32(const void* p) {
    return (unsigned)(size_t)p;
}

__device__ __forceinline__ void wait_async_all() {
    asm volatile("s_wait_asynccnt 0x0" ::: "memory");
}

// ------------------------------- LayerNorm ---------------------------------
__global__ __launch_bounds__(256)
void k_layernorm(const float* __restrict__ x, const float* __restrict__ lw,
                 const float* __restrict__ lb, float* __restrict__ xn) {
    __shared__ float red[8];
    const size_t row = blockIdx.x;
    const float* src = x + row * D_MODEL;
    const int tid = threadIdx.x;
    float vals[3];
    float s = 0.f;
    #pragma unroll
    for (int i = 0; i < 3; ++i) { vals[i] = src[tid + i * 256]; s += vals[i]; }
    #pragma unroll
    for (int o = 16; o > 0; o >>= 1) s += __shfl_xor(s, o, 32);
    if ((tid & 31) == 0) red[tid >> 5] = s;
    __syncthreads();
    float tot = 0.f;
    #pragma unroll
    for (int i = 0; i < 8; ++i) tot += red[i];
    const float mean = tot * (1.f / D_MODEL);
    __syncthreads();
    float v = 0.f;
    #pragma unroll
    for (int i = 0; i < 3; ++i) { float d = vals[i] - mean; v += d * d; }
    #pragma unroll
    for (int o = 16; o > 0; o >>= 1) v += __shfl_xor(v, o, 32);
    if ((tid & 31) == 0) red[tid >> 5] = v;
    __syncthreads();
    float vt = 0.f;
    #pragma unroll
    for (int i = 0; i < 8; ++i) vt += red[i];
    const float inv = rsqrtf(vt * (1.f / D_MODEL) + 1e-5f);
    #pragma unroll
    for (int i = 0; i < 3; ++i) {
        const int c = tid + i * 256;
        xn[row * D_MODEL + c] = (vals[i] - mean) * inv * lw[c] + lb[c];
    }
}

// ----------------------- fp32 WMMA GEMM:  C = A * W^T ----------------------
// A: [M][K] row-major, W: [N][K] row-major (torch-style weight), C: [M][N].
// Block tile 128x64, 8 waves, each wave 32x32 (2x2 accumulators of 16x16).
// K panels of 16 double-buffered in LDS via global_load_async_to_lds_b128;
// inner loop is v_wmma_f32_16x16x4_f32.
#define GBM 128
#define GBN 64
#define GKB 16

__device__ __forceinline__ void stage_panel_async(
        const float* __restrict__ A, const float* __restrict__ W,
        int bm0, int bn0, int K, int k0,
        float* As, float* Bs, int tid) {
    {   // A panel 128x16 floats: 32B per thread, two b128 async transfers
        const int r = tid >> 1;
        const int c = (tid & 1) * 8;
        const float* g = A + (size_t)(bm0 + r) * K + k0 + c;
        const unsigned l = lds_u32(&As[r * GKB + c]);
        asm volatile("global_load_async_to_lds_b128 %0, %1, off\n\t"
                     "global_load_async_to_lds_b128 %0, %1, off offset:16"
                     :: "v"(l), "v"(g) : "memory");
    }
    {   // B panel 64x16 floats: 16B per thread, one b128 async transfer
        const int r = tid >> 2;
        const int c = (tid & 3) * 4;
        const float* g = W + (size_t)(bn0 + r) * K + k0 + c;
        const unsigned l = lds_u32(&Bs[r * GKB + c]);
        asm volatile("global_load_async_to_lds_b128 %0, %1, off"
                     :: "v"(l), "v"(g) : "memory");
    }
}

__global__ __launch_bounds__(256)
void k_gemm_wmma_nt(const float* __restrict__ A, const float* __restrict__ W,
                    float* __restrict__ C, const float* __restrict__ addv,
                    int M, int N, int K, int addmode) {
    __shared__ __align__(16) float As[2][GBM * GKB];   // 2 x 8 KB
    __shared__ __align__(16) float Bs[2][GBN * GKB];   // 2 x 4 KB

    const int tid  = threadIdx.x;
    const int lane = tid & 31;
    const int wave = tid >> 5;          // 0..7
    const int wm   = wave & 3;          // 4 waves along M
    const int wn   = wave >> 2;         // 2 waves along N
    const int bm0  = blockIdx.x * GBM;
    const int bn0  = blockIdx.y * GBN;

    const v8f vzero = {0.f, 0.f, 0.f, 0.f, 0.f, 0.f, 0.f, 0.f};
    v8f acc[2][2];
    #pragma unroll
    for (int i = 0; i < 2; ++i)
        #pragma unroll
        for (int j = 0; j < 2; ++j) acc[i][j] = vzero;

    // ISA 16x4 f32 A/B lane layout: row index = lane&15, K half = lane>>4.
    const int klane = (lane >> 4) * 2;
    const int mrow  = lane & 15;

    const int nPanels = K / GKB;

    // prologue: async-stage panel 0 into buffer 0
    stage_panel_async(A, W, bm0, bn0, K, 0, As[0], Bs[0], tid);
    wait_async_all();
    __syncthreads();

    for (int p = 0; p < nPanels; ++p) {
        const int cur = p & 1;
        const int nxt = cur ^ 1;
        if (p + 1 < nPanels)   // overlap next panel fetch with this panel's WMMAs
            stage_panel_async(A, W, bm0, bn0, K, (p + 1) * GKB,
                              As[nxt], Bs[nxt], tid);

        #pragma unroll
        for (int kk = 0; kk < GKB; kk += 4) {
            v2f afrag[2], bfrag[2];
            #pragma unroll
            for (int i = 0; i < 2; ++i) {
                const int m = wm * 32 + i * 16 + mrow;
                afrag[i] = *(const v2f*)&As[cur][m * GKB + kk + klane];
            }
            #pragma unroll
            for (int j = 0; j < 2; ++j) {
                const int n = wn * 32 + j * 16 + mrow;
                bfrag[j] = *(const v2f*)&Bs[cur][n * GKB + kk + klane];
            }
            #pragma unroll
            for (int i = 0; i < 2; ++i)
                #pragma unroll
                for (int j = 0; j < 2; ++j)
                    acc[i][j] = __builtin_amdgcn_wmma_f32_16x16x4_f32(
                        false, afrag[i], false, bfrag[j],
                        (short)0, acc[i][j], false, false);
        }

        // next buffer must be resident, and all waves must be done reading
        // `cur` before panel p+2's loads (issued next iteration) overwrite it.
        wait_async_all();
        __syncthreads();
    }

    // store: C/D layout — VGPR r: lanes0-15 -> M=r, lanes16-31 -> M=r+8; N=lane&15
    const int nlane = lane & 15;
    const int rsel  = (lane >> 4) * 8;
    #pragma unroll
    for (int i = 0; i < 2; ++i) {
        #pragma unroll
        for (int j = 0; j < 2; ++j) {
            const int mbase = bm0 + wm * 32 + i * 16;
            const int nidx  = bn0 + wn * 32 + j * 16 + nlane;
            #pragma unroll
            for (int r = 0; r < 8; ++r) {
                const int m = mbase + r + rsel;
                const size_t off = (size_t)m * N + nidx;
                float v = acc[i][j][r];
                if (addmode) v += addv[off];
                C[off] = v;
            }
        }
    }
}

// ------------------- depthwise causal conv(4) + bias + SiLU ----------------
__global__ __launch_bounds__(256)
void k_conv_silu(const float* __restrict__ xz, const float* __restrict__ cw,
                 const float* __restrict__ cb, float* __restrict__ xb) {
    const size_t i = (size_t)blockIdx.x * blockDim.x + threadIdx.x;
    const int d = (int)(i % D_INNER);
    const size_t bt = i / D_INNER;
    const int t = (int)(bt % T_LEN);
    const int b = (int)(bt / T_LEN);
    float acc = cb[d];
    #pragma unroll
    for (int j = 0; j < D_CONV; ++j) {
        const int tt = t - (D_CONV - 1) + j;
        if (tt >= 0)
            acc += cw[d * D_CONV + j] *
                   xz[((size_t)b * T_LEN + tt) * (2 * D_INNER) + d];
    }
    xb[i] = acc / (1.f + __expf(-acc));   // silu
}

// --------------------------- x_proj (N = 33) -------------------------------
__global__ __launch_bounds__(256)
void k_xproj(const float* __restrict__ xb, const float* __restrict__ w,
             float* __restrict__ sp) {
    __shared__ __align__(16) float row[D_INNER];
    const size_t bt = blockIdx.x;
    const float* src = xb + bt * D_INNER;
    for (int i = threadIdx.x; i < D_INNER; i += 256) row[i] = src[i];
    __syncthreads();
    if (threadIdx.x < (2 * D_STATE + 1)) {
        const float* wr = w + (size_t)threadIdx.x * D_INNER;
        float acc = 0.f;
        for (int k = 0; k < D_INNER; ++k) acc += row[k] * wr[k];
        sp[bt * SP_STRIDE + threadIdx.x] = acc;
    }
}

// ------------------------------ selective scan -----------------------------
#define SCAN_TS 16
__global__ __launch_bounds__(256)
void k_ssm_scan(const float* __restrict__ sp, const float* __restrict__ xb,
                const float* __restrict__ dtw, const float* __restrict__ dtb,
                const float* __restrict__ A_log, float* __restrict__ y) {
    const int d = blockIdx.x * 256 + threadIdx.x;   // 0..1535
    const int b = blockIdx.y;
    __shared__ float stage[SCAN_TS * 33];

    const float w  = dtw[d];
    const float bb = dtb[d];
    float a[D_STATE], h[D_STATE];
    #pragma unroll
    for (int s = 0; s < D_STATE; ++s) {
        a[s] = -__expf(A_log[d * D_STATE + s]);
        h[s] = 0.f;
    }

    for (int t0 = 0; t0 < T_LEN; t0 += SCAN_TS) {
        __syncthreads();
        // async-stage this chunk's (dt,B,C) rows straight into LDS
        for (int i = threadIdx.x; i < SCAN_TS * 33; i += 256) {
            const int tt = i / 33, e = i % 33;
            const float* g = &sp[((size_t)b * T_LEN + t0 + tt) * SP_STRIDE + e];
            const unsigned l = lds_u32(&stage[i]);
            asm volatile("global_load_async_to_lds_b32 %0, %1, off"
                         :: "v"(l), "v"(g) : "memory");
        }
        wait_async_all();
        __syncthreads();
        for (int tt = 0; tt < SCAN_TS; ++tt) {
            const float* p = &stage[tt * 33];
            const float pre = p[0] * w + bb;
            const float dt = (pre > 20.f) ? pre : log1pf(__expf(pre));
            const size_t idx = ((size_t)b * T_LEN + t0 + tt) * D_INNER + d;
            const float x = xb[idx];
            const float dtx = dt * x;
            float yv = 0.f;
            #pragma unroll
            for (int s = 0; s < D_STATE; ++s) {
                const float hb = __expf(dt * a[s]);
                h[s] = hb * h[s] + dtx * p[1 + s];
                yv += h[s] * p[1 + D_STATE + s];
            }
            y[idx] = yv;
        }
    }
}

// -------------------------- gating epilogue --------------------------------
__global__ __launch_bounds__(256)
void k_gate(const float* __restrict__ xz, const float* __restrict__ xb,
            const float* __restrict__ Dp, float* __restrict__ y) {
    const size_t i = (size_t)blockIdx.x * blockDim.x + threadIdx.x;
    const int d = (int)(i % D_INNER);
    const size_t bt = i / D_INNER;
    const float z = xz[bt * (2 * D_INNER) + D_INNER + d];
    const float sz = z / (1.f + __expf(-z));
    y[i] = (y[i] + Dp[d] * xb[i]) * sz;
}

// ---------------------------------------------------------------------------
extern "C" void kernel_launch(void* const* d_in, const int* in_sizes, int n_in,
                              void* d_out, int out_size, void* d_ws, size_t ws_size,
                              hipStream_t stream) {
    const float* x         = (const float*)d_in[0];
    const float* ln_w      = (const float*)d_in[1];
    const float* ln_b      = (const float*)d_in[2];
    const float* in_proj_w = (const float*)d_in[3];
    const float* conv_w    = (const float*)d_in[4];
    const float* conv_b    = (const float*)d_in[5];
    const float* x_proj_w  = (const float*)d_in[6];
    const float* dt_proj_w = (const float*)d_in[7];
    const float* dt_proj_b = (const float*)d_in[8];
    const float* A_log     = (const float*)d_in[9];
    const float* D_param   = (const float*)d_in[10];
    const float* out_proj_w= (const float*)d_in[11];

    float* ws = (float*)d_ws;
    float* xn = ws;                                   // 8192 x 768
    float* xz = xn + (size_t)BT * D_MODEL;            // 8192 x 3072
    float* xb = xz + (size_t)BT * 2 * D_INNER;        // 8192 x 1536
    float* sp = xb + (size_t)BT * D_INNER;            // 8192 x 40
    float* y  = sp + (size_t)BT * SP_STRIDE;          // 8192 x 1536

    // 1) LayerNorm
    k_layernorm<<<BT, 256, 0, stream>>>(x, ln_w, ln_b, xn);

    // 2) in_proj: xz = xn @ in_proj_w^T   (8192x768 * 768x3072)
    {
        dim3 g(BT / GBM, (2 * D_INNER) / GBN);
        k_gemm_wmma_nt<<<g, 256, 0, stream>>>(xn, in_proj_w, xz, nullptr,
                                              BT, 2 * D_INNER, D_MODEL, 0);
    }

    // 3) causal depthwise conv + bias + silu -> xb
    k_conv_silu<<<((size_t)BT * D_INNER) / 256, 256, 0, stream>>>(xz, conv_w,
                                                                  conv_b, xb);

    // 4) x_proj -> (dt_raw, B, C) padded rows
    k_xproj<<<BT, 256, 0, stream>>>(xb, x_proj_w, sp);

    // 5) selective scan
    {
        dim3 g(D_INNER / 256, B_SZ);
        k_ssm_scan<<<g, 256, 0, stream>>>(sp, xb, dt_proj_w, dt_proj_b,
                                          A_log, y);
    }

    // 6) y = (y + D*xb) * silu(z)
    k_gate<<<((size_t)BT * D_INNER) / 256, 256, 0, stream>>>(xz, xb, D_param, y);

    // 7) out_proj + residual: out = x + y @ out_proj_w^T  (8192x1536 * 1536x768)
    {
        dim3 g(BT / GBM, D_MODEL / GBN);
        k_gemm_wmma_nt<<<g, 256, 0, stream>>>(y, out_proj_w, (float*)d_out, x,
                                              BT, D_MODEL, D_INNER, 1);
    }
}